// MemoryModule_84000970375258
// MI455X (gfx1250) — compile-verified
//
#include <hip/hip_runtime.h>

typedef float v2f __attribute__((ext_vector_type(2)));
typedef float v4f __attribute__((ext_vector_type(4)));
typedef float v8f __attribute__((ext_vector_type(8)));
typedef unsigned int v4u __attribute__((ext_vector_type(4)));
typedef int v4i __attribute__((ext_vector_type(4)));
typedef int v8i __attribute__((ext_vector_type(8)));

#define WPAD 132            // padded LDS row stride (floats): conflict-free fragment reads
#define SMEM_FLOATS (2 * 128 * WPAD)
#define SMEM_BYTES (SMEM_FLOATS * 4)

#if defined(__HIP_DEVICE_COMPILE__) && __has_builtin(__builtin_amdgcn_tensor_load_to_lds)
#define HAVE_TDM 1
#else
#define HAVE_TDM 0
#endif

// Block constant-folding of an LDS byte offset into DS-instruction immediates:
// keeps one address VGPR per tile so the 16-bit DS offset field suffices.
static __device__ __forceinline__ unsigned opaque_u32(unsigned x) {
#if defined(__HIP_DEVICE_COMPILE__)
    asm("" : "+v"(x));
#endif
    return x;
}

// F32 WMMA: D[16x16] = A[16x4] * B[4x16] + C (wave32; 2 VGPRs per A/B, 8 per C/D).
static __device__ __forceinline__ v8f wmma_f32_16x16x4(v2f a, v2f b, v8f c) {
#if defined(__HIP_DEVICE_COMPILE__) && __has_builtin(__builtin_amdgcn_wmma_f32_16x16x4_f32)
    return __builtin_amdgcn_wmma_f32_16x16x4_f32(
        /*neg_a=*/false, a, /*neg_b=*/false, b,
        /*c_mod=*/(short)0, c, /*reuse_a=*/false, /*reuse_b=*/false);
#else
    (void)a; (void)b;
    return c;  // host-pass stub, never executed
#endif
}

static __device__ __forceinline__ float sigmoidf_dev(float x) {
    return 1.0f / (1.0f + __expf(-x));
}

#if HAVE_TDM
// TDM: DMA a 128x128 f32 tile (row-major, stride 128) from global into LDS at
// lds_byte_off, padding +4 DWORDs after every 128 DWORDs (-> WPAD=132 rows).
// D# packing per CDNA5 ISA §8.3/8.4:
//  g0: [1:0]=count=1 | [63:32]=lds_addr | [120:64]=global_addr | [127:126]=type=2
//  g1: [17:16]=data_size=2(4B) | [20]=pad_en | [24:22]=pad_interval=6(128dw)
//      | [31:25]=pad_amount=3(4dw) | tensor_dim0/1=128 | tile_dim0/1=128
//      | tensor_dim0_stride=128
static __device__ __forceinline__ void tdm_load_tile_128x128(const float* gsrc,
                                                            unsigned lds_byte_off) {
    const unsigned long long ga = (unsigned long long)(uintptr_t)gsrc;
    v4u g0 = {1u,                                  // count=1
              lds_byte_off,                        // lds_addr (bytes)
              (unsigned)(ga & 0xFFFFFFFFu),        // global_addr[31:0]
              ((unsigned)((ga >> 32) & 0x01FFFFFFu)) | 0x80000000u};  // [56:32] | type=2
    v8i g1 = {(int)0x07920000,   // data_size=4B, pad_en, interval=128dw, amount=4dw
              (int)(128u << 16), // tensor_dim0 = 128 (bits 79:48)
              (int)(128u << 16), // tensor_dim1 = 128 (bits 111:80)
              (int)(128u << 16), // tile_dim0   = 128 (bits 127:112)
              (int)128,          // tile_dim1   = 128 (bits 143:128), tile_dim2=0
              (int)128,          // tensor_dim0_stride = 128 (bits 207:160)
              0, 0};             // tensor_dim1_stride = 0 (unused for 2D tile)
    v4i z4 = {0, 0, 0, 0};
#if __clang_major__ >= 23
    v8i z8 = {0, 0, 0, 0, 0, 0, 0, 0};
    __builtin_amdgcn_tensor_load_to_lds(g0, g1, z4, z4, z8, 0);
#else
    __builtin_amdgcn_tensor_load_to_lds(g0, g1, z4, z4, 0);
#endif
}
#endif

static __device__ __forceinline__ void wait_tensorcnt0() {
#if defined(__HIP_DEVICE_COMPILE__)
#if __has_builtin(__builtin_amdgcn_s_wait_tensorcnt)
    __builtin_amdgcn_s_wait_tensorcnt(0);
#else
    asm volatile("s_wait_tensorcnt 0x0" ::: "memory");
#endif
#endif
}

// Streaming copy of memory -> out (N*H floats), 16B vectors + non-temporal hints.
__global__ void memcpy_nt_kernel(const v4f* __restrict__ src,
                                 v4f* __restrict__ dst, long long n4) {
    long long i = (long long)blockIdx.x * blockDim.x + threadIdx.x;
    long long stride = (long long)gridDim.x * blockDim.x;
    for (; i < n4; i += stride) {
        v4f v = __builtin_nontemporal_load(&src[i]);
        __builtin_nontemporal_store(v, &dst[i]);
    }
}

// One wave handles 16 batch rows; H = 128. Weights staged per-gate in LDS via TDM.
// Gate order (PyTorch GRUCell): rows [0,128)=r, [128,256)=z, [256,384)=n.
__global__ __launch_bounds__(256) void gru_update_kernel(
    const int* __restrict__ node_ids,
    const float* __restrict__ messages,   // [B,128]
    const float* __restrict__ memory,     // [N,128]
    const float* __restrict__ W_ih,       // [384,128]
    const float* __restrict__ W_hh,       // [384,128]
    const float* __restrict__ b_ih,       // [384]
    const float* __restrict__ b_hh,       // [384]
    float* __restrict__ out,              // [N,128]
    int B) {
    extern __shared__ float smem[];  // [2][128][WPAD]

    const int lane = threadIdx.x & 31;
    const int wave = threadIdx.x >> 5;

    // Clamp OOB waves to the last tile instead of returning: every wave must
    // reach the barriers; duplicated waves recompute and rewrite identical
    // values (deterministic, benign).
    long long tile = (long long)blockIdx.x * 8 + wave;
    const long long last_tile = ((long long)B + 15) / 16 - 1;
    if (tile > last_tile) tile = last_tile;
    const long long base = tile * 16;

    const int halfsel = lane >> 4;   // 0: lanes 0-15, 1: lanes 16-31
    const int rloc = lane & 15;
    const int kb = halfsel * 2;      // K-pair offset inside each 16x4 fragment

    long long row = base + rloc;
    if (row >= B) row = B - 1;

    // ---- preload A fragments for X (messages) and H (gathered memory) ----
    const float* xrow = messages + (size_t)row * 128;
    const int hrow = node_ids[row];
    const float* hprow = memory + (size_t)hrow * 128;

    v2f xf[32], hf[32];
#pragma unroll
    for (int k = 0; k < 32; ++k) {
        const int off = 4 * k + kb;
        xf[k] = *(const v2f*)(xrow + off);
        hf[k] = *(const v2f*)(hprow + off);
    }

    // Scatter targets for C/D-layout elements (VGPR e -> batch row base+e+8*halfsel)
    int srows[8];
#pragma unroll
    for (int e = 0; e < 8; ++e) {
        long long grow = base + e + halfsel * 8;
        if (grow >= B) grow = B - 1;
        srows[e] = node_ids[grow];
    }

    v8f rt[8], zt[8];

#pragma unroll
    for (int gate = 0; gate < 3; ++gate) {
        __syncthreads();  // previous gate's LDS reads done before overwrite
#if HAVE_TDM
        // ---- TDM stage: W_ih[gate] -> smem[0], W_hh[gate] -> smem[1] ----
        if (wave == 0) {  // once per workgroup; TDM ignores EXEC
            const unsigned lds0 = (unsigned)(uintptr_t)smem;  // low 32 bits = LDS addr
            tdm_load_tile_128x128(W_ih + (size_t)gate * 128 * 128, lds0);
            tdm_load_tile_128x128(W_hh + (size_t)gate * 128 * 128,
                                  lds0 + 128 * WPAD * 4);
            wait_tensorcnt0();
        }
#else
        // ---- manual stage (batched: all loads in flight, then stores) ----
#pragma unroll
        for (int m = 0; m < 2; ++m) {
            const float* src = (m ? W_hh : W_ih) + (size_t)gate * 128 * 128;
            float* dst = smem + m * 128 * WPAD;
            v4f tmp[16];
#pragma unroll
            for (int i = 0; i < 16; ++i) {
                const int flat = threadIdx.x + 256 * i;  // v4f index, 0..4095
                tmp[i] = *(const v4f*)(src + (flat >> 5) * 128 + (flat & 31) * 4);
            }
#pragma unroll
            for (int i = 0; i < 16; ++i) {
                const int flat = threadIdx.x + 256 * i;
                *(v4f*)(dst + (flat >> 5) * WPAD + (flat & 31) * 4) = tmp[i];
            }
        }
#endif
        __syncthreads();

        // ---- 8 feature tiles of 16 columns each ----
#pragma unroll
        for (int c = 0; c < 8; ++c) {
            const int feat = c * 16 + rloc;  // output feature owned by this lane

            // Opaque per-tile byte offsets -> one DS address register each,
            // inner-loop offsets (16k <= 0x1F0) live in the DS offset field.
            const unsigned offi =
                opaque_u32(((unsigned)(c * 16 + rloc) * WPAD + (unsigned)kb) * 4u);
            const unsigned offh = opaque_u32(offi + 128u * WPAD * 4u);
            const float* lwi = (const float*)((const char*)smem + offi);
            const float* lwh = (const float*)((const char*)smem + offh);

            v8f ai = {0, 0, 0, 0, 0, 0, 0, 0};
            v8f ah = {0, 0, 0, 0, 0, 0, 0, 0};
#pragma unroll
            for (int k = 0; k < 32; ++k) {
                v2f bi = *(const v2f*)(lwi + 4 * k);
                v2f bh = *(const v2f*)(lwh + 4 * k);
                ai = wmma_f32_16x16x4(xf[k], bi, ai);
                ah = wmma_f32_16x16x4(hf[k], bh, ah);
            }

            if (gate == 0) {  // r
                const float br = b_ih[feat] + b_hh[feat];
#pragma unroll
                for (int e = 0; e < 8; ++e)
                    rt[c][e] = sigmoidf_dev(ai[e] + ah[e] + br);
            } else if (gate == 1) {  // z
                const float bz = b_ih[128 + feat] + b_hh[128 + feat];
#pragma unroll
                for (int e = 0; e < 8; ++e)
                    zt[c][e] = sigmoidf_dev(ai[e] + ah[e] + bz);
            } else {  // n + combine + scatter
                const float bin = b_ih[256 + feat];
                const float bhn = b_hh[256 + feat];
#pragma unroll
                for (int e = 0; e < 8; ++e) {
                    const size_t off = (size_t)srows[e] * 128 + feat;
                    const float hval = memory[off];
                    const float nn =
                        tanhf(ai[e] + bin + rt[c][e] * (ah[e] + bhn));
                    const float res = (1.0f - zt[c][e]) * nn + zt[c][e] * hval;
                    out[off] = res;
                }
            }
        }
    }
}

extern "C" void kernel_launch(void* const* d_in, const int* in_sizes, int n_in,
                              void* d_out, int out_size, void* d_ws, size_t ws_size,
                              hipStream_t stream) {
    const int* node_ids   = (const int*)d_in[0];
    const float* messages = (const float*)d_in[1];
    const float* memory   = (const float*)d_in[2];
    const float* W_ih     = (const float*)d_in[3];
    const float* W_hh     = (const float*)d_in[4];
    const float* b_ih     = (const float*)d_in[5];
    const float* b_hh     = (const float*)d_in[6];
    float* out = (float*)d_out;

    const int B = in_sizes[0];                    // 100000
    const long long NH = (long long)in_sizes[2];  // N * 128

    // 1) copy memory -> out (untouched rows; updated rows overwritten below)
    const long long n4 = NH / 4;
    long long cg = (n4 + 255) / 256;
    if (cg > 131072) cg = 131072;
    if (cg < 1) cg = 1;
    memcpy_nt_kernel<<<dim3((unsigned)cg), dim3(256), 0, stream>>>(
        (const v4f*)memory, (v4f*)out, n4);

    // 2) GRU update + scatter (one wave per 16 rows, 8 waves per block,
    //    135 KB dynamic LDS for per-gate weight staging via TDM)
    const int tiles = (B + 15) / 16;
    const int blocks = (tiles + 7) / 8;
    gru_update_kernel<<<dim3(blocks), dim3(256), SMEM_BYTES, stream>>>(
        node_ids, messages, memory, W_ih, W_hh, b_ih, b_hh, out, B);
}